// XAMIRoIHeads_8117488190273
// MI455X (gfx1250) — compile-verified
//
#include <hip/hip_runtime.h>
#include <cstdint>

#define B_IMG 8
#define N_BOX 4096
#define C_CLS 91
#define M_CAND (N_BOX * (C_CLS - 1))   // 368640 candidates per image
#define DETS 100
#define CHUNK 4096                      // floats per TDM staging chunk (16 KB)
#define NCHUNK (M_CAND / CHUNK)         // 90

typedef __attribute__((ext_vector_type(2))) float v2f;
typedef __attribute__((ext_vector_type(8))) float v8f;
typedef __attribute__((ext_vector_type(4))) unsigned int u32x4;
typedef __attribute__((ext_vector_type(8))) int i32x8;
typedef __attribute__((ext_vector_type(4))) int i32x4;

// ---------------------------------------------------------------------------
// Tensor Data Mover: DMA a contiguous run of f32 from global into LDS.
// D# built per CDNA5 ISA ch.8: 2-D tensor (tile = one row of nElems f32).
// ---------------------------------------------------------------------------
__device__ __forceinline__ void tdm_load_f32_chunk(const float* gsrc,
                                                   unsigned lds_byte_off,
                                                   int nElems) {
  unsigned long long ga = (unsigned long long)(uintptr_t)gsrc;
  u32x4 g0;
  g0.x = 1u;                                              // count=1, no gather
  g0.y = lds_byte_off;                                    // lds_addr
  g0.z = (unsigned)ga;                                    // global_addr[31:0]
  g0.w = (unsigned)((ga >> 32) & 0x01ffffffull) | (2u << 30); // addr[56:32] | type=2
  i32x8 g1;
  g1[0] = 0x00020000;                                     // wg_mask=0, data_size=4B
  g1[1] = (int)((unsigned)nElems << 16);                  // tensor_dim0 lo16
  g1[2] = (int)((((unsigned)nElems >> 16) & 0xffffu) | (1u << 16)); // dim0 hi | dim1=1
  g1[3] = (int)(((unsigned)nElems & 0xffffu) << 16);      // tile_dim0
  g1[4] = 1;                                              // tile_dim1=1
  g1[5] = nElems;                                         // dim0_stride lo32
  g1[6] = 0;
  g1[7] = 0;
  i32x4 z4 = {0, 0, 0, 0};
#if __has_include(<hip/amd_detail/amd_gfx1250_TDM.h>)
  // amdgpu-toolchain (therock) 6-arg form
  i32x8 z8 = {0, 0, 0, 0, 0, 0, 0, 0};
  __builtin_amdgcn_tensor_load_to_lds(g0, g1, z4, z4, z8, 0);
#else
  // ROCm 7.2 5-arg form
  __builtin_amdgcn_tensor_load_to_lds(g0, g1, z4, z4, 0);
#endif
}

// ---------------------------------------------------------------------------
// Kernel A: softmax (WMMA row sums) + box decode + clip + threshold.
// 256 threads = 8 waves; each wave owns 16 boxes (rows). Grid never straddles
// an image (4096 % 128 == 0), so EXEC is full at every WMMA.
// ---------------------------------------------------------------------------
__global__ __launch_bounds__(256) void score_decode_kernel(
    const float* __restrict__ logits,     // [B*N, 91]
    const float* __restrict__ regress,    // [B*N, 364]
    const float* __restrict__ proposals,  // [B*N, 4]
    const int* __restrict__ shapes,       // [B, 2] (H, W)
    float* __restrict__ run,              // [B, M]
    float* __restrict__ boxes)            // [B, M, 4]
{
  __shared__ float eLds[8][16][92];
  __shared__ float sumLds[8][16];

  const int w = threadIdx.x >> 5;
  const int ln = threadIdx.x & 31;
  const int m = ln & 15;        // row within wave tile
  const int half = ln >> 4;     // K-half per ISA 16x4 f32 A layout
  const long rowBase = (long)blockIdx.x * 128 + (long)w * 16;
  const long rowG = rowBase + m;
  const float* lrow = logits + rowG * C_CLS;

  // row max over 91 classes: lane pair (l, l+16) splits 46/45, combine via xor-16
  float mx = -3.4e38f;
  {
    const int c0 = half * 46;
    const int c1 = half ? C_CLS : 46;
    for (int c = c0; c < c1; ++c) mx = fmaxf(mx, lrow[c]);
  }
  mx = fmaxf(mx, __shfl_xor(mx, 16, 32));

  // exp + WMMA row-sum: A = 16x4 exp chunk (lane l holds rows m=l&15, K=2*half,2*half+1),
  // B = ones 4x16  =>  D[m][*] accumulates sum_k exp.
  v8f acc = {0.f, 0.f, 0.f, 0.f, 0.f, 0.f, 0.f, 0.f};
  v2f ones; ones.x = 1.0f; ones.y = 1.0f;
  for (int base = 0; base < 92; base += 4) {
    const int k0 = base + 2 * half;
    const float e0 = (k0 < C_CLS) ? expf(lrow[k0] - mx) : 0.0f;
    const float e1 = (k0 + 1 < C_CLS) ? expf(lrow[k0 + 1] - mx) : 0.0f;
    eLds[w][m][k0] = e0;
    eLds[w][m][k0 + 1] = e1;
    v2f a; a.x = e0; a.y = e1;
    acc = __builtin_amdgcn_wmma_f32_16x16x4_f32(
        false, a, false, ones, (short)0, acc, false, false);
  }
  // D layout: lane l, VGPR r -> row (r + 8*(l>>4)); every column identical (B=ones).
  if (m == 0) {
#pragma unroll
    for (int r = 0; r < 8; ++r) sumLds[w][half * 8 + r] = acc[r];
  }
  __syncthreads();

  // decode + clip + threshold for this wave's 16 rows x 90 fg classes (45 iters/lane)
  const int bImg = (int)((long)blockIdx.x * 128 / N_BOX);
  const float Hf = (float)shapes[bImg * 2 + 0];
  const float Wf = (float)shapes[bImg * 2 + 1];
  const float CLIPV = 4.135166556742356f;  // log(1000/16)
  for (int idx = ln; idx < 16 * 90; idx += 32) {
    const int mm = idx / 90;
    const int cc = idx - mm * 90 + 1;      // class 1..90
    const long row = rowBase + mm;
    const int n = (int)(row - (long)bImg * N_BOX);
    const float score = eLds[w][mm][cc] / sumLds[w][mm];
    const float4 p = reinterpret_cast<const float4*>(proposals)[row];
    const float4 r4 = reinterpret_cast<const float4*>(regress + row * (C_CLS * 4))[cc];
    const float bw0 = p.z - p.x, bh0 = p.w - p.y;
    const float cx = p.x + 0.5f * bw0, cy = p.y + 0.5f * bh0;
    const float dx = r4.x / 10.0f, dy = r4.y / 10.0f;
    const float dw = fminf(r4.z / 5.0f, CLIPV), dh = fminf(r4.w / 5.0f, CLIPV);
    const float pcx = dx * bw0 + cx, pcy = dy * bh0 + cy;
    const float pw = expf(dw) * bw0, ph = expf(dh) * bh0;
    const float x1 = fminf(fmaxf(pcx - 0.5f * pw, 0.0f), Wf);
    const float y1 = fminf(fmaxf(pcy - 0.5f * ph, 0.0f), Hf);
    const float x2 = fminf(fmaxf(pcx + 0.5f * pw, 0.0f), Wf);
    const float y2 = fminf(fmaxf(pcy + 0.5f * ph, 0.0f), Hf);
    const bool keep = (score > 0.05f) && ((x2 - x1) >= 0.01f) && ((y2 - y1) >= 0.01f);
    const size_t g = (size_t)bImg * M_CAND + (size_t)n * 90 + (size_t)(cc - 1);
    run[g] = keep ? score : -1.0f;
    reinterpret_cast<float4*>(boxes)[g] = make_float4(x1, y1, x2, y2);
  }
}

// ---------------------------------------------------------------------------
// Kernel B: sequential NMS, one block (32 waves) per image.
// argmax pass scans `run` from LDS, staged by TDM with double buffering.
// Per-class offset trick == "same label else IoU=0" (class slabs are disjoint).
// ---------------------------------------------------------------------------
__global__ __launch_bounds__(1024) void nms_kernel(
    float* __restrict__ run, const float* __restrict__ boxes,
    float* __restrict__ outB, float* __restrict__ outS, int* __restrict__ outL)
{
  __shared__ float stage[2][CHUNK];  // 32 KB double buffer (written by TDM)
  __shared__ float rv[32];
  __shared__ int ri[32];
  __shared__ float sV;
  __shared__ int sI;

  const int b = blockIdx.x;
  const int tid = threadIdx.x;
  const int wv = tid >> 5, ln = tid & 31;
  float* runI = run + (size_t)b * M_CAND;
  const float4* boxI = reinterpret_cast<const float4*>(boxes) + (size_t)b * M_CAND;
  const bool issuer = (wv == 0);  // wave-uniform: only wave 0 drives the TDM
  const unsigned ldsOff0 = (unsigned)(uintptr_t)&stage[0][0];
  const unsigned ldsOff1 = (unsigned)(uintptr_t)&stage[1][0];

  // Anchor the staging buffer with real program stores so its ds_loads can
  // never be folded as loads of a never-written shared global; the TDM
  // overwrites this immediately.
  for (int e = 0; e < 4; ++e) {
    stage[0][tid * 4 + e] = -1.0f;
    stage[1][tid * 4 + e] = -1.0f;
  }
  __syncthreads();

  for (int t = 0; t < DETS; ++t) {
    // ---- argmax over run[0..M), first-index tie-break (== jnp.argmax) ----
    float bv = -2.0f;
    int bi = 0x7fffffff;
    if (issuer) tdm_load_f32_chunk(runI, ldsOff0, CHUNK);
    for (int c = 0; c < NCHUNK; ++c) {
      if (issuer) {
        if (c + 1 < NCHUNK) {
          tdm_load_f32_chunk(runI + (size_t)(c + 1) * CHUNK,
                             (c & 1) ? ldsOff0 : ldsOff1, CHUNK);
          __builtin_amdgcn_s_wait_tensorcnt(1);  // chunk c complete (in-order)
        } else {
          __builtin_amdgcn_s_wait_tensorcnt(0);
        }
        asm volatile("" ::: "memory");  // no value forwarding across the wait
      }
      __syncthreads();  // chunk c visible in LDS to all waves
      const float* sp = &stage[c & 1][tid * 4];
      const int jb = c * CHUNK + tid * 4;
#pragma unroll
      for (int e = 0; e < 4; ++e) {
        const float v = sp[e];
        if (v > bv) { bv = v; bi = jb + e; }  // ascending j keeps earliest on tie
      }
      __syncthreads();  // reads done before buffer is re-targeted
    }
#pragma unroll
    for (int d = 16; d > 0; d >>= 1) {
      const float ov = __shfl_xor(bv, d, 32);
      const int oi = __shfl_xor(bi, d, 32);
      if (ov > bv || (ov == bv && oi < bi)) { bv = ov; bi = oi; }
    }
    if (ln == 0) { rv[wv] = bv; ri[wv] = bi; }
    __syncthreads();
    if (wv == 0) {
      bv = rv[ln]; bi = ri[ln];
#pragma unroll
      for (int d = 16; d > 0; d >>= 1) {
        const float ov = __shfl_xor(bv, d, 32);
        const int oi = __shfl_xor(bi, d, 32);
        if (ov > bv || (ov == bv && oi < bi)) { bv = ov; bi = oi; }
      }
      if (ln == 0) { sV = bv; sI = bi; }
    }
    __syncthreads();
    const float s = sV;
    const int i = sI;

    if (s <= 0.0f) {  // uniform: nothing left; remaining outputs are zeros
      for (int tt = t; tt < DETS; ++tt) {
        const int o = b * DETS + tt;
        if (tid < 4) outB[o * 4 + tid] = 0.0f;
        else if (tid == 4) outS[o] = 0.0f;
        else if (tid == 5) outL[o] = 0;
      }
      break;
    }

    const float4 wb = boxI[i];
    const int wl = i % 90;
    if (tid == 0) {
      const int o = b * DETS + t;
      outB[o * 4 + 0] = wb.x; outB[o * 4 + 1] = wb.y;
      outB[o * 4 + 2] = wb.z; outB[o * 4 + 3] = wb.w;
      outS[o] = s;
      outL[o] = wl + 1;
    }

    // ---- suppress: same-class IoU > 0.5 (includes the winner itself) ----
    const float wa = (wb.z - wb.x) * (wb.w - wb.y);
    for (int j = tid; j < M_CAND; j += 1024) {
      const float v = runI[j];
      if (v > 0.0f && (j % 90) == wl) {
        const float4 bb = boxI[j];
        const float lx = fmaxf(wb.x, bb.x), ly = fmaxf(wb.y, bb.y);
        const float rx = fminf(wb.z, bb.z), ry = fminf(wb.w, bb.w);
        const float iw = fmaxf(rx - lx, 0.0f), ih = fmaxf(ry - ly, 0.0f);
        const float inter = iw * ih;
        const float a2 = (bb.z - bb.x) * (bb.w - bb.y);
        const float iou = inter / (wa + a2 - inter);
        if (iou > 0.5f) runI[j] = -1.0f;
      }
    }
    __threadfence();   // make suppression stores L2-visible to the TDM
    __syncthreads();
  }
}

// ---------------------------------------------------------------------------
extern "C" void kernel_launch(void* const* d_in, const int* in_sizes, int n_in,
                              void* d_out, int out_size, void* d_ws, size_t ws_size,
                              hipStream_t stream) {
  const float* logits = (const float*)d_in[0];     // [8,4096,91]
  const float* regress = (const float*)d_in[1];    // [8,4096,364]
  const float* proposals = (const float*)d_in[2];  // [8,4096,4]
  const int* shapes = (const int*)d_in[3];         // [8,2]

  float* run = (float*)d_ws;                           // [8, M] f32
  float* boxes = run + (size_t)B_IMG * M_CAND;         // [8, M, 4] f32

  float* outB = (float*)d_out;                         // [8,100,4]
  float* outS = outB + (size_t)B_IMG * DETS * 4;       // [8,100]
  int* outL = (int*)(outB + (size_t)B_IMG * DETS * 5); // [8,100] int32

  score_decode_kernel<<<dim3((B_IMG * N_BOX) / 128), dim3(256), 0, stream>>>(
      logits, regress, proposals, shapes, run, boxes);
  nms_kernel<<<dim3(B_IMG), dim3(1024), 0, stream>>>(run, boxes, outB, outS, outL);

  (void)in_sizes; (void)n_in; (void)out_size; (void)ws_size;
}